// GNNLeakDetectorTopo_52682068852861
// MI455X (gfx1250) — compile-verified
//
#include <hip/hip_runtime.h>
#include <hip/hip_bf16.h>
#include <math.h>

typedef __attribute__((ext_vector_type(2))) float v2f;
typedef __attribute__((ext_vector_type(8))) float v8f;

// ---------------------------------------------------------------------------
// Degree / normalization kernels
// ---------------------------------------------------------------------------
__global__ void k_deg_init(float* deg, int n) {
    int i = blockIdx.x * blockDim.x + threadIdx.x;
    if (i < n) deg[i] = 1.0f;  // self-loop contributes 1
}

__global__ void k_deg_scatter(const int* __restrict__ dst, float* deg, int e) {
    int i = blockIdx.x * blockDim.x + threadIdx.x;
    if (i < e) atomicAdd(&deg[dst[i]], 1.0f);
}

__global__ void k_deg_rsqrt(float* deg, int n) {
    int i = blockIdx.x * blockDim.x + threadIdx.x;
    if (i < n) deg[i] = rsqrtf(deg[i]);  // deg >= 1 always (self-loop)
}

// ---------------------------------------------------------------------------
// Topo MLP (16 -> 32 -> 32, ReLU) + concat with x into h [N x 160]
// ---------------------------------------------------------------------------
__global__ __launch_bounds__(128) void k_topo_concat(
    const float* __restrict__ x, const float* __restrict__ topo,
    const float* __restrict__ tW1, const float* __restrict__ tb1,
    const float* __restrict__ tW2, const float* __restrict__ tb2,
    float* __restrict__ h, int n)
{
    int i = blockIdx.x * blockDim.x + threadIdx.x;
    if (i >= n) return;

    float t[16];
#pragma unroll
    for (int j = 0; j < 16; ++j) t[j] = topo[(size_t)i * 16 + j];

    float h1[32];
#pragma unroll
    for (int o = 0; o < 32; ++o) {
        float s = tb1[o];
#pragma unroll
        for (int j = 0; j < 16; ++j) s = fmaf(t[j], tW1[j * 32 + o], s);
        h1[o] = fmaxf(s, 0.0f);
    }

    // copy x part (vectorized float4)
    const float4* xs = (const float4*)(x + (size_t)i * 128);
    float4* hd = (float4*)(h + (size_t)i * 160);
#pragma unroll
    for (int j = 0; j < 32; ++j) hd[j] = xs[j];

#pragma unroll
    for (int o = 0; o < 32; ++o) {
        float s = tb2[o];
#pragma unroll
        for (int j = 0; j < 32; ++j) s = fmaf(h1[j], tW2[j * 32 + o], s);
        h[(size_t)i * 160 + 128 + o] = fmaxf(s, 0.0f);
    }
}

// ---------------------------------------------------------------------------
// WMMA fp32 GEMM:  Out[N x 64] = H[N x K] * W[K x 64]   (optionally relu(H))
// One wave -> 16x64 output tile (4 x v8f accumulators), K-step 4.
//
// W staged in LDS repacked as K-pairs with 160-float pitch:
//   sW[(k/2)*160 + col*2 + (k&1)] = W[k][col]
// -> each B fragment {W[k][col], W[k+1][col]} is one aligned 8-byte word
//    (single conflict-free ds_load_b64 straight into an even VGPR pair:
//     khalf=0 lanes hit banks 0-31, khalf=1 lanes hit banks 32-63).
// ---------------------------------------------------------------------------
template <int K, bool RELU_A>
__global__ __launch_bounds__(256) void k_gemm_wmma(
    const float* __restrict__ H, const float* __restrict__ W,
    float* __restrict__ Out, int n)
{
    constexpr int PITCH = 160;  // floats per k-pair row (pad 128 -> 160 for banks)
    __shared__ float sW[(K / 2) * PITCH];
    for (int i = threadIdx.x; i < K * 64; i += 256) {
        int k = i >> 6, col = i & 63;
        sW[(k >> 1) * PITCH + col * 2 + (k & 1)] = W[i];
    }
    __syncthreads();

    const int wave  = threadIdx.x >> 5;
    const int lane  = threadIdx.x & 31;
    const int rlane = lane & 15;
    const int khalf = lane >> 4;           // 0: K+0..1, 1: K+2..3
    const int row0  = (blockIdx.x * 8 + wave) * 16;
    if (row0 >= n) return;                 // uniform per-wave exit; EXEC all-ones below

    v8f acc0 = {}, acc1 = {}, acc2 = {}, acc3 = {};
    const float* hrow = H + (size_t)(row0 + rlane) * K;

    for (int k = 0; k < K; k += 4) {
        v2f a = *(const v2f*)(hrow + k + 2 * khalf);     // aligned b64 load
        if (RELU_A) { a.x = fmaxf(a.x, 0.0f); a.y = fmaxf(a.y, 0.0f); }

        const v2f* bp = (const v2f*)&sW[(k / 2 + khalf) * PITCH];
        v2f b0 = bp[rlane];
        v2f b1 = bp[rlane + 16];
        v2f b2 = bp[rlane + 32];
        v2f b3 = bp[rlane + 48];

        acc0 = __builtin_amdgcn_wmma_f32_16x16x4_f32(false, a, false, b0, (short)0, acc0, false, false);
        acc1 = __builtin_amdgcn_wmma_f32_16x16x4_f32(false, a, false, b1, (short)0, acc1, false, false);
        acc2 = __builtin_amdgcn_wmma_f32_16x16x4_f32(false, a, false, b2, (short)0, acc2, false, false);
        acc3 = __builtin_amdgcn_wmma_f32_16x16x4_f32(false, a, false, b3, (short)0, acc3, false, false);
    }

    // C/D layout: VGPR i -> row (row0 + i + 8*khalf), col = colgroup*16 + rlane
    float* obase = Out + (size_t)(row0 + 8 * khalf) * 64 + rlane;
#pragma unroll
    for (int i = 0; i < 8; ++i) {
        float* orow = obase + (size_t)i * 64;
        orow[0]  = acc0[i];
        orow[16] = acc1[i];
        orow[32] = acc2[i];
        orow[48] = acc3[i];
    }
}

// ---------------------------------------------------------------------------
// GCN aggregation: out = selfloop + bias, then atomic edge scatter
// ---------------------------------------------------------------------------
__global__ void k_self_init(const float* __restrict__ hw, const float* __restrict__ dis,
                            const float* __restrict__ bias, float* __restrict__ out, int n)
{
    int idx = blockIdx.x * blockDim.x + threadIdx.x;
    if (idx >= n * 64) return;
    int i = idx >> 6, j = idx & 63;
    float d = dis[i];
    out[idx] = fmaf(hw[idx], d * d, bias[j]);
}

// 16 lanes per edge, float4 gather + 4 scalar f32 atomics; prefetch next edge
__global__ void k_edge_scatter(const int* __restrict__ src, const int* __restrict__ dst,
                               const float* __restrict__ hw, const float* __restrict__ dis,
                               float* out, int e)
{
    int t = blockIdx.x * blockDim.x + threadIdx.x;
    int edge = t >> 4;
    if (edge >= e) return;
    int q = (t & 15) * 4;
    int s = src[edge], d = dst[edge];

    // prefetch the row we'll gather one grid-stride ahead (global_prefetch_b8)
    int nedge = edge + (gridDim.x * blockDim.x >> 4);
    if (nedge < e) __builtin_prefetch(hw + (size_t)src[nedge] * 64 + q, 0, 1);

    float norm = dis[s] * dis[d];
    float4 v = *(const float4*)(hw + (size_t)s * 64 + q);
    float* o = out + (size_t)d * 64 + q;
    atomicAdd(o + 0, v.x * norm);
    atomicAdd(o + 1, v.y * norm);
    atomicAdd(o + 2, v.z * norm);
    atomicAdd(o + 3, v.w * norm);
}

// ---------------------------------------------------------------------------
// Head: sigmoid(relu(h2) . oW + ob)
// ---------------------------------------------------------------------------
__global__ void k_head(const float* __restrict__ h2, const float* __restrict__ oW,
                       const float* __restrict__ ob, float* __restrict__ out, int n)
{
    int i = blockIdx.x * blockDim.x + threadIdx.x;
    if (i >= n) return;
    float s = ob[0];
    const float* row = h2 + (size_t)i * 64;
#pragma unroll
    for (int j = 0; j < 64; ++j) s = fmaf(fmaxf(row[j], 0.0f), oW[j], s);
    out[i] = 1.0f / (1.0f + __expf(-s));
}

// ---------------------------------------------------------------------------
// Launcher
// ---------------------------------------------------------------------------
extern "C" void kernel_launch(void* const* d_in, const int* in_sizes, int n_in,
                              void* d_out, int out_size, void* d_ws, size_t ws_size,
                              hipStream_t stream) {
    const float* x    = (const float*)d_in[0];
    const float* topo = (const float*)d_in[1];
    const int*   eidx = (const int*)d_in[2];
    const float* tW1  = (const float*)d_in[3];
    const float* tb1  = (const float*)d_in[4];
    const float* tW2  = (const float*)d_in[5];
    const float* tb2  = (const float*)d_in[6];
    const float* cW1  = (const float*)d_in[7];
    const float* cb1  = (const float*)d_in[8];
    const float* cW2  = (const float*)d_in[9];
    const float* cb2  = (const float*)d_in[10];
    const float* oW   = (const float*)d_in[11];
    const float* ob   = (const float*)d_in[12];

    const int N = in_sizes[0] / 128;
    const int E = in_sizes[2] / 2;
    const int* src = eidx;
    const int* dst = eidx + E;

    float* ws   = (float*)d_ws;
    float* h    = ws;                         // N*160
    float* bufA = h + (size_t)N * 160;        // N*64  (GEMM output)
    float* bufB = bufA + (size_t)N * 64;      // N*64  (aggregation accumulator)
    float* dis  = bufB + (size_t)N * 64;      // N     (deg -> deg^-1/2 in place)

    const int TPB = 256;
    const int rowTiles = (N + 15) / 16;
    const int gemmBlocks = (rowTiles + 7) / 8;
    const int scatterBlocks = (int)((((size_t)E * 16) + TPB - 1) / TPB);

    // degree -> deg^-1/2
    k_deg_init<<<(N + TPB - 1) / TPB, TPB, 0, stream>>>(dis, N);
    k_deg_scatter<<<(E + TPB - 1) / TPB, TPB, 0, stream>>>(dst, dis, E);
    k_deg_rsqrt<<<(N + TPB - 1) / TPB, TPB, 0, stream>>>(dis, N);

    // topo MLP + concat
    k_topo_concat<<<(N + 127) / 128, 128, 0, stream>>>(x, topo, tW1, tb1, tW2, tb2, h, N);

    // layer 1: GEMM (K=160) -> self-loop+bias -> edge scatter
    k_gemm_wmma<160, false><<<gemmBlocks, TPB, 0, stream>>>(h, cW1, bufA, N);
    k_self_init<<<((size_t)N * 64 + TPB - 1) / TPB, TPB, 0, stream>>>(bufA, dis, cb1, bufB, N);
    k_edge_scatter<<<scatterBlocks, TPB, 0, stream>>>(src, dst, bufA, dis, bufB, E);

    // layer 2: GEMM (K=64, ReLU fused on A load) -> self-loop+bias -> edge scatter
    k_gemm_wmma<64, true><<<gemmBlocks, TPB, 0, stream>>>(bufB, cW2, bufA, N);
    k_self_init<<<((size_t)N * 64 + TPB - 1) / TPB, TPB, 0, stream>>>(bufA, dis, cb2, bufB, N);
    k_edge_scatter<<<scatterBlocks, TPB, 0, stream>>>(src, dst, bufA, dis, bufB, E);

    // head (layer-2 ReLU fused here)
    k_head<<<(N + TPB - 1) / TPB, TPB, 0, stream>>>(bufB, oW, ob, (float*)d_out, N);
}